// Routing_10402410791507
// MI455X (gfx1250) — compile-verified
//
#include <hip/hip_runtime.h>
#include <hip/hip_bf16.h>

typedef float v2f __attribute__((ext_vector_type(2)));
typedef float v4f __attribute__((ext_vector_type(4)));
typedef float v8f __attribute__((ext_vector_type(8)));

#define B_T 16      // batch
#define NI  1024    // n_inp
#define NJ  64      // n_out
#define DC  4       // d_cov
#define DH  16      // d_out
#define CH  64      // DC*DH
#define EPSF 1e-5f
#define I_CHUNK 16
#define NCHUNK (NI / I_CHUNK)   // 64

// workspace layout (float offsets)
#define WS_ACCA   0                       // [16][64]  a_out accumulator
#define WS_ACCSD  1024                    // [16][64]  sum D_use
#define WS_ACCSV  2048                    // [16][64][4][16] sum D*V
#define WS_ACCSV2 67584                   // [16][64][4][16] sum D*V^2
#define WS_MUT    133120                  // [4bg][64][32lane][8v] tile-swizzled mu
#define WS_ISIGT  198656                  // [4bg][64][32lane][8v] tile-swizzled 1/(2 sig2)
#define WS_LOGDET 264192                  // [16][64]  0.5*sum log sig2
#define WS_LSA    265216                  // [16][64]  log_sigmoid(a_out)
#define WS_WF     266240                  // [1024*64][32][8] fragment-swizzled W
#define WS_BF     17043456                // [1024*64][32][4] fragment-swizzled Bias
#define N_ACC     133120

__global__ __launch_bounds__(256) void zero_acc_kernel(float* __restrict__ ws) {
  int idx = blockIdx.x * 256 + threadIdx.x;
  if (idx < N_ACC) ws[idx] = 0.0f;
}

// One-time fragment swizzle of W and Bias into WMMA lane order.
// grid: 8192 blocks x 256 threads; wave w of block handles pair p = blk*8+w.
__global__ __launch_bounds__(256) void swizzle_wb_kernel(
    const float* __restrict__ Wt,    // [1024*64][16][16]  (d,h)
    const float* __restrict__ Bias,  // [1024*64][4][16]   (c,h)
    float* __restrict__ ws)
{
  const int tid  = threadIdx.x;
  const int wid  = tid >> 5;
  const int lane = tid & 31;
  const int hi   = lane >> 4;
  const int n    = lane & 15;
  const int p    = blockIdx.x * 8 + wid;   // (i*NJ + j), 0..65535

  const float* wp = Wt + (size_t)p * 256;
  v8f wv;
#pragma unroll
  for (int k = 0; k < 4; ++k) {
    wv[2 * k + 0] = wp[(4 * k + 2 * hi + 0) * 16 + n];
    wv[2 * k + 1] = wp[(4 * k + 2 * hi + 1) * 16 + n];
  }
  float* wfp = ws + WS_WF + ((size_t)p * 32 + lane) * 8;
  *(v4f*)(wfp + 0) = v4f{wv[0], wv[1], wv[2], wv[3]};
  *(v4f*)(wfp + 4) = v4f{wv[4], wv[5], wv[6], wv[7]};

  const float* bp = Bias + (size_t)p * CH;
  v4f bf;
#pragma unroll
  for (int c = 0; c < 4; ++c) bf[c] = bp[c * 16 + n];
  *(v4f*)(ws + WS_BF + ((size_t)p * 32 + lane) * 4) = bf;
}

// grid: (NCHUNK, 4 b-groups); block: 256 (8 waves). Wave w owns j = 8w..8w+7.
__global__ __launch_bounds__(256) void routing_pass_kernel(
    const float* __restrict__ a_inp,     // [16][1024]
    const float* __restrict__ mu_inp,    // [16][1024][4][16]
    const float* __restrict__ beta_use,  // [1024][64]
    const float* __restrict__ beta_ign,  // [1024][64]
    float* __restrict__ ws,
    int iter)
{
  __shared__ float lds_logp[NJ][4];   // log-likelihood per (j, b_local)
  __shared__ float lds_max[4];
  __shared__ float lds_inv[4];
  __shared__ float lds_SD[NJ][4];     // chunk-local sum D_use
  __shared__ float lds_aacc[NJ][4];   // chunk-local a_out contribution
  __shared__ float lds_asig[4];       // sigmoid(a_inp[b][i]) for the 4 b's

  const int tid  = threadIdx.x;
  const int wid  = tid >> 5;          // wave 0..7
  const int lane = tid & 31;
  const int hi   = lane >> 4;         // half-wave
  const int ln16 = lane & 15;
  const int bg   = blockIdx.y;        // b-group (4 batches)
  const int i0   = blockIdx.x * I_CHUNK;

  // A-fragment source row for this lane: M = ln16 = 4*bl + c
  const int a_b  = bg * 4 + (ln16 >> 2);
  const int a_c  = ln16 & 3;

  // zero chunk-local LDS accumulators
  {
    int jj = tid >> 2, bl = tid & 3;
    lds_SD[jj][bl]   = 0.0f;
    lds_aacc[jj][bl] = 0.0f;
  }

  // persistent register accumulators: one v8f tile per owned j
  v8f acc_sv[8], acc_sv2[8];
#pragma unroll
  for (int jj = 0; jj < 8; ++jj)
#pragma unroll
    for (int v = 0; v < 8; ++v) { acc_sv[jj][v] = 0.0f; acc_sv2[jj][v] = 0.0f; }

  for (int i = i0; i < i0 + I_CHUNK; ++i) {
    if (tid < 4)
      lds_asig[tid] = 1.0f / (1.0f + __expf(-a_inp[(bg * 4 + tid) * NI + i]));

    // ---- A fragments straight from global (two consecutive d's = b64 load) ----
    const float* ap = mu_inp + ((size_t)(a_b * NI + i) * DC + a_c) * 16 + 2 * hi;
    v2f afr[4];
#pragma unroll
    for (int k = 0; k < 4; ++k) afr[k] = *(const v2f*)(ap + 4 * k);

    // ---- compute V tiles for the 8 owned j's; log-likelihoods if iter>0 ----
    v8f vt[8];
#pragma unroll
    for (int jj = 0; jj < 8; ++jj) {
      const int j = wid * 8 + jj;
      const size_t p = (size_t)(i * NJ + j);
      const float* wfp = ws + WS_WF + (p * 32 + lane) * 8;
      v4f w0 = *(const v4f*)(wfp + 0);
      v4f w1 = *(const v4f*)(wfp + 4);
      // prefetch next i's swizzled W for this j (speculative; dropped if OOB)
      __builtin_prefetch(wfp + (size_t)NJ * 256, 0, 1);

      v4f bf = *(const v4f*)(ws + WS_BF + (p * 32 + lane) * 4);
      v8f cacc;
      cacc[0] = bf[0]; cacc[1] = bf[1]; cacc[2] = bf[2]; cacc[3] = bf[3];
      cacc[4] = bf[0]; cacc[5] = bf[1]; cacc[6] = bf[2]; cacc[7] = bf[3];

      v2f b0; b0.x = w0[0]; b0.y = w0[1];
      v2f b1; b1.x = w0[2]; b1.y = w0[3];
      v2f b2; b2.x = w1[0]; b2.y = w1[1];
      v2f b3; b3.x = w1[2]; b3.y = w1[3];
      cacc = __builtin_amdgcn_wmma_f32_16x16x4_f32(false, afr[0], false, b0, (short)0, cacc, false, false);
      cacc = __builtin_amdgcn_wmma_f32_16x16x4_f32(false, afr[1], false, b1, (short)0, cacc, false, false);
      cacc = __builtin_amdgcn_wmma_f32_16x16x4_f32(false, afr[2], false, b2, (short)0, cacc, false, false);
      cacc = __builtin_amdgcn_wmma_f32_16x16x4_f32(false, afr[3], false, b3, (short)0, cacc, false, false);
      vt[jj] = cacc;

      if (iter > 0) {
        // (V - mu)^2 * 1/(2 sig2), tile-layout loads (b128 x4)
        const size_t tbase = ((size_t)(bg * NJ + j) * 32 + lane) * 8;
        v4f m0 = *(const v4f*)(ws + WS_MUT + tbase);
        v4f m1 = *(const v4f*)(ws + WS_MUT + tbase + 4);
        v4f s0v = *(const v4f*)(ws + WS_ISIGT + tbase);
        v4f s1v = *(const v4f*)(ws + WS_ISIGT + tbase + 4);
        float s0 = 0.0f, s1 = 0.0f;
#pragma unroll
        for (int v = 0; v < 4; ++v) {
          float dm = cacc[v] - m0[v];
          s0 += dm * dm * s0v[v];
        }
#pragma unroll
        for (int v = 0; v < 4; ++v) {
          float dm = cacc[v + 4] - m1[v];
          s1 += dm * dm * s1v[v];
        }
        // reduce across the 16 lanes of each half-wave
#pragma unroll
        for (int m = 1; m < 16; m <<= 1) {
          s0 += __shfl_xor(s0, m, 32);
          s1 += __shfl_xor(s1, m, 32);
        }
        if (ln16 == 0) {
          int bl = 2 * hi, b = bg * 4 + bl;
          lds_logp[j][bl] = ws[WS_LSA + b * NJ + j] - s0 - ws[WS_LOGDET + b * NJ + j];
        }
        if (ln16 == 1) {
          int bl = 2 * hi + 1, b = bg * 4 + bl;
          lds_logp[j][bl] = ws[WS_LSA + b * NJ + j] - s1 - ws[WS_LOGDET + b * NJ + j];
        }
      }
    }
    __syncthreads();

    // ---- softmax over j per b_local: 64 threads, half-wave tree reductions ----
    if (iter > 0 && tid < 64) {
      const int bl = tid >> 4;        // each 16-lane group handles one b_local
      const int g  = tid & 15;
      float l0 = lds_logp[4 * g + 0][bl];
      float l1 = lds_logp[4 * g + 1][bl];
      float l2 = lds_logp[4 * g + 2][bl];
      float l3 = lds_logp[4 * g + 3][bl];
      float mx = fmaxf(fmaxf(l0, l1), fmaxf(l2, l3));
#pragma unroll
      for (int m = 1; m < 16; m <<= 1) mx = fmaxf(mx, __shfl_xor(mx, m, 32));
      float s = __expf(l0 - mx) + __expf(l1 - mx) + __expf(l2 - mx) + __expf(l3 - mx);
#pragma unroll
      for (int m = 1; m < 16; m <<= 1) s += __shfl_xor(s, m, 32);
      if (g == 0) { lds_max[bl] = mx; lds_inv[bl] = 1.0f / s; }
    }
    __syncthreads();

    // ---- accumulate statistics ----
#pragma unroll
    for (int jj = 0; jj < 8; ++jj) {
      const int j = wid * 8 + jj;
      float du[4];
#pragma unroll
      for (int bl = 0; bl < 4; ++bl) {
        float R = (iter == 0)
                    ? (1.0f / NJ)
                    : __expf(lds_logp[j][bl] - lds_max[bl]) * lds_inv[bl];
        du[bl] = lds_asig[bl] * R;
      }
      if (lane == 0) {
        float bu = beta_use[i * NJ + j];
        float bi = beta_ign[i * NJ + j];
#pragma unroll
        for (int bl = 0; bl < 4; ++bl) {
          lds_SD[j][bl]   += du[bl];
          lds_aacc[j][bl] += bu * du[bl] - bi * (lds_asig[bl] - du[bl]);
        }
      }
      float duA = hi ? du[2] : du[0];   // rows v<4
      float duB = hi ? du[3] : du[1];   // rows v>=4
#pragma unroll
      for (int v = 0; v < 8; ++v) {
        float dsel = (v < 4) ? duA : duB;
        float val  = vt[jj][v];
        float dv   = dsel * val;
        acc_sv[jj][v]  += dv;
        acc_sv2[jj][v] += dv * val;
      }
    }
    __syncthreads();
  }

  // ---- flush chunk accumulators to global (float atomics) ----
#pragma unroll
  for (int jj = 0; jj < 8; ++jj) {
    const int j = wid * 8 + jj;
#pragma unroll
    for (int v = 0; v < 8; ++v) {
      int row = v + 8 * hi;
      int b   = bg * 4 + (row >> 2);
      int c   = v & 3;
      int idx = ((b * NJ + j) * DC + c) * DH + ln16;
      atomicAdd(&ws[WS_ACCSV  + idx], acc_sv[jj][v]);
      atomicAdd(&ws[WS_ACCSV2 + idx], acc_sv2[jj][v]);
    }
  }
  {
    int j = tid >> 2, bl = tid & 3;
    int b = bg * 4 + bl;
    atomicAdd(&ws[WS_ACCSD + b * NJ + j], lds_SD[j][bl]);
    atomicAdd(&ws[WS_ACCA  + b * NJ + j], lds_aacc[j][bl]);
  }
}

// grid: (NJ, B); block: 64 threads (one per (c,h))
__global__ __launch_bounds__(64) void routing_finalize_kernel(
    float* __restrict__ ws, float* __restrict__ out, int write_out)
{
  const int j  = blockIdx.x;
  const int b  = blockIdx.y;
  const int t  = threadIdx.x;       // c*16 + h
  const int bj = b * NJ + j;

  float SD   = ws[WS_ACCSD + bj];
  float over = 1.0f / (SD + EPSF);
  int idx    = bj * CH + t;
  float sv   = ws[WS_ACCSV + idx];
  float sv2  = ws[WS_ACCSV2 + idx];
  float mu   = over * sv;
  float sig  = over * (sv2 - 2.0f * mu * sv + mu * mu * SD) + EPSF;

  // store mu and 1/(2 sig2) in WMMA-tile-swizzled order for the pass kernel
  {
    const int c   = t >> 4, h = t & 15;
    const int bgi = b >> 2, bl = b & 3;
    const int ln  = (bl >> 1) * 16 + h;        // lane in wave
    const int v   = (bl & 1) * 4 + c;          // slot in D tile
    const size_t ti = ((size_t)(bgi * NJ + j) * 32 + ln) * 8 + v;
    ws[WS_MUT + ti]   = mu;
    ws[WS_ISIGT + ti] = 0.5f / sig;
  }

  // logdet = 0.5 * sum_ch log(sig)
  float l = 0.5f * __logf(sig);
#pragma unroll
  for (int m = 1; m < 32; m <<= 1) l += __shfl_xor(l, m, 32);
  __shared__ float red[2];
  if ((t & 31) == 0) red[t >> 5] = l;
  __syncthreads();

  if (t == 0) {
    ws[WS_LOGDET + bj] = red[0] + red[1];
    float a = ws[WS_ACCA + bj];
    ws[WS_LSA + bj] = fminf(a, 0.0f) - log1pf(__expf(-fabsf(a)));  // log_sigmoid
    if (write_out) out[bj] = a;                    // a_out
  }
  if (write_out) {
    out[1024 + idx]         = mu;                  // mu_out
    out[1024 + 65536 + idx] = sig;                 // sig2
  }
}

extern "C" void kernel_launch(void* const* d_in, const int* in_sizes, int n_in,
                              void* d_out, int out_size, void* d_ws, size_t ws_size,
                              hipStream_t stream) {
  const float* a_inp   = (const float*)d_in[0];
  const float* mu_inp  = (const float*)d_in[1];
  const float* Wt      = (const float*)d_in[2];
  const float* Bias    = (const float*)d_in[3];
  const float* b_use   = (const float*)d_in[4];
  const float* b_ign   = (const float*)d_in[5];
  float* out = (float*)d_out;
  float* ws  = (float*)d_ws;

  swizzle_wb_kernel<<<8192, 256, 0, stream>>>(Wt, Bias, ws);

  for (int it = 0; it < 3; ++it) {
    zero_acc_kernel<<<(N_ACC + 255) / 256, 256, 0, stream>>>(ws);
    routing_pass_kernel<<<dim3(NCHUNK, 4), 256, 0, stream>>>(
        a_inp, mu_inp, b_use, b_ign, ws, it);
    routing_finalize_kernel<<<dim3(NJ, B_T), 64, 0, stream>>>(
        ws, out, it == 2 ? 1 : 0);
  }
}